// PathAdd_40003325395149
// MI455X (gfx1250) — compile-verified
//
#include <hip/hip_runtime.h>
#include <stdint.h>

#define D_FEAT 128
#define WAVE 32

// SADDR-form hardware f32 atomic add, device scope:
//   global_atomic_add_f32 voffset, vdata, s[base:base+1] offset:imm scope:SCOPE_DEV
// Uniform 64-bit base in SGPRs, per-lane 32-bit byte offset in a VGPR.
#define GATOMIC_ADD_F32(base_sgpr, voff, val, OFFSTR)                          \
  asm volatile("global_atomic_add_f32 %0, %1, %2 offset:" OFFSTR               \
               " scope:SCOPE_DEV"                                              \
               :                                                               \
               : "v"(voff), "v"(val), "s"(base_sgpr)                           \
               : "memory")

// ---------------------------------------------------------------------------
// Kernel 1: zero the accumulator (d_out) and the per-node degree array (d_ws)
// ---------------------------------------------------------------------------
__global__ void gnn_zero_kernel(float* __restrict__ out, int* __restrict__ deg,
                                int n_nodes) {
  const int tid    = blockIdx.x * blockDim.x + threadIdx.x;
  const int stride = gridDim.x * blockDim.x;
  const int nvec   = n_nodes * (D_FEAT / 4);
  float4 z = make_float4(0.f, 0.f, 0.f, 0.f);
  float4* o4 = reinterpret_cast<float4*>(out);
  for (int i = tid; i < nvec; i += stride) o4[i] = z;
  for (int i = tid; i < n_nodes; i += stride) deg[i] = 0;
}

// ---------------------------------------------------------------------------
// Kernel 2: one wave32 per edge.
//   edge id is forced wave-uniform via readfirstlane so src/dst come in as
//   scalar (SMEM) loads and row bases live in SGPRs. Each lane then does one
//   global_load_b128 of the source row (512 B per wave, coalesced) and four
//   SADDR-form global_atomic_add_f32 into the destination row.
// ---------------------------------------------------------------------------
__global__ void gnn_scatter_kernel(const float* __restrict__ feats,
                                   const int*   __restrict__ src,
                                   const int*   __restrict__ dst,
                                   float*       __restrict__ out,
                                   int*         __restrict__ deg,
                                   int n_edges) {
  const int gtid = blockIdx.x * blockDim.x + threadIdx.x;
  const int lane = threadIdx.x & 31;
  // wave-uniform edge index, made visible to the compiler as uniform
  const int edge = __builtin_amdgcn_readfirstlane(gtid >> 5);
  if (edge >= n_edges) return;

  const int s = src[edge];   // scalar load (uniform index)
  const int d = dst[edge];   // scalar load (uniform index)

  // Source row: uniform base + per-lane float4 offset -> global_load_b128
  const float4* srow =
      reinterpret_cast<const float4*>(feats + (size_t)s * D_FEAT);
  float4 v = srow[lane];

  // Destination row: uniform 64-bit base (SGPR pair) + lane byte offset
  const uint64_t obase =
      (uint64_t)(uintptr_t)(out + (size_t)d * D_FEAT);
  const uint32_t voff = (uint32_t)(lane * 16);

  GATOMIC_ADD_F32(obase, voff, v.x, "0");
  GATOMIC_ADD_F32(obase, voff, v.y, "4");
  GATOMIC_ADD_F32(obase, voff, v.z, "8");
  GATOMIC_ADD_F32(obase, voff, v.w, "12");

  if (lane == 0) atomicAdd(deg + d, 1);
}

// ---------------------------------------------------------------------------
// Kernel 3: one wave32 per node; zero-in-degree nodes keep their own feats.
// ---------------------------------------------------------------------------
__global__ void gnn_finalize_kernel(const float* __restrict__ feats,
                                    const int*   __restrict__ deg,
                                    float*       __restrict__ out,
                                    int n_nodes) {
  const int gtid = blockIdx.x * blockDim.x + threadIdx.x;
  const int lane = threadIdx.x & 31;
  const int node = __builtin_amdgcn_readfirstlane(gtid >> 5);
  if (node >= n_nodes) return;

  if (deg[node] == 0) {   // scalar load (uniform index)
    const float4* srow =
        reinterpret_cast<const float4*>(feats + (size_t)node * D_FEAT);
    float4* orow = reinterpret_cast<float4*>(out + (size_t)node * D_FEAT);
    orow[lane] = srow[lane];
  }
}

// ---------------------------------------------------------------------------
extern "C" void kernel_launch(void* const* d_in, const int* in_sizes, int n_in,
                              void* d_out, int out_size, void* d_ws,
                              size_t ws_size, hipStream_t stream) {
  (void)n_in; (void)out_size; (void)ws_size;

  const float* feats = reinterpret_cast<const float*>(d_in[0]);
  const int*   src   = reinterpret_cast<const int*>(d_in[1]);
  const int*   dst   = reinterpret_cast<const int*>(d_in[2]);
  float*       out   = reinterpret_cast<float*>(d_out);
  int*         deg   = reinterpret_cast<int*>(d_ws);   // n_nodes ints

  const int n_nodes = in_sizes[0] / D_FEAT;            // 100000
  const int n_edges = in_sizes[1];                     // 1600000

  const int block = 256;

  // 1) zero accumulator + degree
  gnn_zero_kernel<<<1024, block, 0, stream>>>(out, deg, n_nodes);

  // 2) scatter-add: one wave per edge -> n_edges*32 threads
  {
    long long threads = (long long)n_edges * WAVE;
    int grid = (int)((threads + block - 1) / block);
    gnn_scatter_kernel<<<grid, block, 0, stream>>>(feats, src, dst, out, deg,
                                                   n_edges);
  }

  // 3) finalize zero-degree nodes: one wave per node
  {
    long long threads = (long long)n_nodes * WAVE;
    int grid = (int)((threads + block - 1) / block);
    gnn_finalize_kernel<<<grid, block, 0, stream>>>(feats, deg, out, n_nodes);
  }
}